// SimplePIDSE_44530220925580
// MI455X (gfx1250) — compile-verified
//
#include <hip/hip_runtime.h>
#include <hip/hip_bf16.h>
#include <math.h>

typedef __attribute__((ext_vector_type(2))) float v2f;
typedef __attribute__((ext_vector_type(8))) float v8f;

#define DT 0.01f

// ---------------------------------------------------------------------------
// Branch-free tanh on the hardware transcendental path:
//   tanh(x) = (e^{2x} - 1) * rcp(e^{2x} + 1)
// v_med3 + v_exp_f32 + v_rcp_f32: straight-line VALU, no divergence.
// ---------------------------------------------------------------------------
__device__ __forceinline__ float fast_tanh(float x) {
    x = fminf(fmaxf(x, -10.f), 10.f);                   // v_med3, avoids inf/inf
    float e = __expf(2.f * x);                          // v_exp_f32 path
    return (e - 1.f) * __builtin_amdgcn_rcpf(e + 1.f);  // v_rcp_f32
}

// ---------------------------------------------------------------------------
// Main kernel: one wave32 per block, 16 EKF chains per wave.
//
// MLP layers run as fp32 WMMA (v_wmma_f32_16x16x4_f32) over the 16-row tile.
//
// EKF structure exploit: with H=I, Q=0.01*I, R=0.1*I, P0=0.1*I the covariance
// stays an exact scalar multiple of the identity for every chain:
//     Pp = (p+0.01) I ;  S = (p+0.11) I ;  K = k I,  k = (p+0.01)/(p+0.11)
//     P' = (1-k)(p+0.01) I
// (matrix inverse of a diagonal matrix keeps off-diagonals exactly zero, so
// this matches the reference math including in floating point up to the
// rounding of the one scalar division per step). The per-chain update is
// just s <- pred + k*(z - pred).
// ---------------------------------------------------------------------------
__global__ __launch_bounds__(32) void pinn_ekf_scan(
    const float* __restrict__ gStates, const float* __restrict__ gControls,
    const float* __restrict__ gMeas,
    const float* __restrict__ gW1, const float* __restrict__ gb1,
    const float* __restrict__ gW2, const float* __restrict__ gb2,
    const float* __restrict__ gW3, const float* __restrict__ gb3,
    float* __restrict__ est, float* __restrict__ partials, int T) {

    __shared__ float sW1[8][32];    // rows 6,7 zero-padded
    __shared__ float sW2[32][16];
    __shared__ float sW3[16][16];   // cols 4..15 zero-padded
    __shared__ float sb1[32];
    __shared__ float sb2[16];
    __shared__ float sb3[16];       // entries 4..15 zero
    __shared__ float X [16][8];     // layer-1 input tile (K padded to 8)
    __shared__ float H1[16][32];
    __shared__ float H2[16][16];
    __shared__ float PO[16][4];     // pinn output (x0.1 applied)
    __shared__ float red[32];

    const int lane = threadIdx.x;        // 0..31, wave32
    const int m    = lane & 15;          // tile row (batch within tile)
    const int half = lane >> 4;          // 0 | 1
    const int kb   = half * 2;           // A/B K-offset within a K=4 chunk
    const int n    = lane & 15;          // B/C column

    // ---- load weights into LDS (zero-padded) ----
    for (int i = lane; i < 8 * 32; i += 32) {
        int r = i >> 5, c = i & 31;
        sW1[r][c] = (r < 6) ? gW1[r * 32 + c] : 0.f;
    }
    for (int i = lane; i < 32 * 16; i += 32) sW2[i >> 4][i & 15] = gW2[i];
    for (int i = lane; i < 16 * 16; i += 32) {
        int r = i >> 4, c = i & 15;
        sW3[r][c] = (c < 4) ? gW3[r * 4 + c] : 0.f;
    }
    sb1[lane] = gb1[lane];
    if (lane < 16) {
        sb2[lane] = gb2[lane];
        sb3[lane] = (lane < 4) ? gb3[lane] : 0.f;
        X[lane][6] = 0.f; X[lane][7] = 0.f;   // permanent K-padding
    }

    const int    b  = blockIdx.x * 16 + m;   // global chain id (valid for m-lanes)
    const size_t bT = (size_t)b * (size_t)T;
    float lacc = 0.f;
    float u0 = 0.f, u1 = 0.f;                // u_prev for the next step
    float s0 = 0.f, s1 = 0.f, s2 = 0.f, s3 = 0.f;  // EKF state, lane-m resident

    // Scalar covariance recursion (uniform across all lanes/chains).
    float p = 0.1f;

    // ---- t = 0: EKF update from raw state ----
    {
        float pp = p + 0.01f;
        float k  = pp / (pp + 0.1f);         // one exact scalar division
        p = (1.f - k) * pp;
        if (lane < 16) {
            const float4 sv = *reinterpret_cast<const float4*>(gStates + bT * 4);
            const float4 zv = *reinterpret_cast<const float4*>(gMeas + bT * 4);
            s0 = sv.x + k * (zv.x - sv.x);
            s1 = sv.y + k * (zv.y - sv.y);
            s2 = sv.z + k * (zv.z - sv.z);
            s3 = sv.w + k * (zv.w - sv.w);
            float4 ov = {s0, s1, s2, s3};
            *reinterpret_cast<float4*>(est + bT * 4) = ov;
            float d0 = s0 - sv.x, d1 = s1 - sv.y, d2 = s2 - sv.z, d3 = s3 - sv.w;
            lacc += d0 * d0 + d1 * d1 + d2 * d2 + d3 * d3;
            // stage layer-1 input tile for t = 1
            const float2 uv = *reinterpret_cast<const float2*>(gControls + bT * 2);
            u0 = uv.x; u1 = uv.y;
            X[lane][0] = s0; X[lane][1] = s1; X[lane][2] = s2; X[lane][3] = s3;
            X[lane][4] = u0; X[lane][5] = u1;
        }
    }

    // ---- sequential scan over time ----
    for (int t = 1; t < T; ++t) {
        __syncthreads();   // X (and on first iter: weights) visible to all lanes

        // Phase B: MLP via fp32 WMMA (uniform control flow, EXEC all-1s)
        // Layer 1: [16x8] x [8x32] -> H1 (two N-tiles, two K=4 chunks each)
#pragma unroll
        for (int nt = 0; nt < 2; ++nt) {
            v8f acc;
#pragma unroll
            for (int r = 0; r < 8; ++r) acc[r] = sb1[nt * 16 + n];
#pragma unroll
            for (int kc = 0; kc < 2; ++kc) {
                v2f a, bm;
                a[0]  = X[m][kc * 4 + kb + 0];
                a[1]  = X[m][kc * 4 + kb + 1];
                bm[0] = sW1[kc * 4 + kb + 0][nt * 16 + n];
                bm[1] = sW1[kc * 4 + kb + 1][nt * 16 + n];
                acc = __builtin_amdgcn_wmma_f32_16x16x4_f32(
                    false, a, false, bm, (short)0, acc, false, false);
            }
#pragma unroll
            for (int r = 0; r < 8; ++r)
                H1[r + 8 * half][nt * 16 + n] = fast_tanh(acc[r]);
        }
        __syncthreads();

        // Layer 2: [16x32] x [32x16] -> H2 (eight K=4 chunks)
        {
            v8f acc;
#pragma unroll
            for (int r = 0; r < 8; ++r) acc[r] = sb2[n];
#pragma unroll
            for (int kc = 0; kc < 8; ++kc) {
                v2f a, bm;
                a[0]  = H1[m][kc * 4 + kb + 0];
                a[1]  = H1[m][kc * 4 + kb + 1];
                bm[0] = sW2[kc * 4 + kb + 0][n];
                bm[1] = sW2[kc * 4 + kb + 1][n];
                acc = __builtin_amdgcn_wmma_f32_16x16x4_f32(
                    false, a, false, bm, (short)0, acc, false, false);
            }
#pragma unroll
            for (int r = 0; r < 8; ++r)
                H2[r + 8 * half][n] = fast_tanh(acc[r]);
        }
        __syncthreads();

        // Layer 3: [16x16] x [16x16(padN)] -> PO (four K=4 chunks), scaled 0.1
        {
            v8f acc;
#pragma unroll
            for (int r = 0; r < 8; ++r) acc[r] = sb3[n];
#pragma unroll
            for (int kc = 0; kc < 4; ++kc) {
                v2f a, bm;
                a[0]  = H2[m][kc * 4 + kb + 0];
                a[1]  = H2[m][kc * 4 + kb + 1];
                bm[0] = sW3[kc * 4 + kb + 0][n];
                bm[1] = sW3[kc * 4 + kb + 1][n];
                acc = __builtin_amdgcn_wmma_f32_16x16x4_f32(
                    false, a, false, bm, (short)0, acc, false, false);
            }
            if (n < 4) {
#pragma unroll
                for (int r = 0; r < 8; ++r) PO[r + 8 * half][n] = acc[r] * 0.1f;
            }
        }
        __syncthreads();

        // Phase C: scalar-gain EKF update + stage next input tile
        float pp = p + 0.01f;
        float k  = pp / (pp + 0.1f);         // uniform scalar division
        p = (1.f - k) * pp;
        if (lane < 16) {
            float pr0 = s0 + s2 * DT + PO[lane][0];
            float pr1 = s1 + s3 * DT + PO[lane][1];
            float pr2 = s2 + u0 * DT + PO[lane][2];
            float pr3 = s3 + u1 * DT + PO[lane][3];

            const float4 zv = *reinterpret_cast<const float4*>(
                gMeas + (bT + (size_t)t) * 4);
            s0 = pr0 + k * (zv.x - pr0);
            s1 = pr1 + k * (zv.y - pr1);
            s2 = pr2 + k * (zv.z - pr2);
            s3 = pr3 + k * (zv.w - pr3);

            float4 ov = {s0, s1, s2, s3};
            *reinterpret_cast<float4*>(est + (bT + (size_t)t) * 4) = ov;

            const float4 sv = *reinterpret_cast<const float4*>(
                gStates + (bT + (size_t)t) * 4);
            float d0 = s0 - sv.x, d1 = s1 - sv.y, d2 = s2 - sv.z, d3 = s3 - sv.w;
            lacc += d0 * d0 + d1 * d1 + d2 * d2 + d3 * d3;

            // stage layer-1 input tile for step t+1 (u_prev = controls[t];
            // in-bounds even at t = T-1, harmless extra staging there)
            const float2 uv = *reinterpret_cast<const float2*>(
                gControls + (bT + (size_t)t) * 2);
            u0 = uv.x; u1 = uv.y;
            X[lane][0] = s0; X[lane][1] = s1; X[lane][2] = s2; X[lane][3] = s3;
            X[lane][4] = u0; X[lane][5] = u1;
        }
    }

    // deterministic per-block partial sum (fixed order)
    __syncthreads();
    red[lane] = lacc;
    __syncthreads();
    if (lane == 0) {
        float tot = 0.f;
#pragma unroll
        for (int i = 0; i < 32; ++i) tot += red[i];
        partials[blockIdx.x] = tot;
    }
}

// Fixed-order final reduction: deterministic across graph replays (no atomics).
__global__ __launch_bounds__(32) void loss_reduce(const float* __restrict__ partials,
                                                  int n, float* __restrict__ out,
                                                  float inv_count) {
    const int lane = threadIdx.x;
    float s = 0.f;
    for (int i = lane; i < n; i += 32) s += partials[i];
    __shared__ float red[32];
    red[lane] = s;
    __syncthreads();
    if (lane == 0) {
        float tot = 0.f;
        for (int i = 0; i < 32; ++i) tot += red[i];
        *out = tot * inv_count;
    }
}

extern "C" void kernel_launch(void* const* d_in, const int* in_sizes, int n_in,
                              void* d_out, int out_size, void* d_ws, size_t ws_size,
                              hipStream_t stream) {
    const float* states   = (const float*)d_in[0];
    const float* controls = (const float*)d_in[1];
    const float* meas     = (const float*)d_in[2];
    const float* W1 = (const float*)d_in[3];
    const float* b1 = (const float*)d_in[4];
    const float* W2 = (const float*)d_in[5];
    const float* b2 = (const float*)d_in[6];
    const float* W3 = (const float*)d_in[7];
    const float* b3 = (const float*)d_in[8];

    const int T  = 512;                       // reference shape
    const int BT = in_sizes[0] / 4;           // B*T
    const int B  = BT / T;                    // 8192
    const int nblocks = B / 16;               // 16 chains per wave32

    float* est      = (float*)d_out;          // [B,T,4] followed by scalar loss
    float* loss_ptr = est + (size_t)BT * 4;
    float* partials = (float*)d_ws;           // nblocks floats of scratch

    pinn_ekf_scan<<<nblocks, 32, 0, stream>>>(states, controls, meas,
                                              W1, b1, W2, b2, W3, b3,
                                              est, partials, T);
    loss_reduce<<<1, 32, 0, stream>>>(partials, nblocks, loss_ptr,
                                      1.0f / ((float)BT * 4.0f));
}